// QuantumLayer_41377714929693
// MI455X (gfx1250) — compile-verified
//
#include <hip/hip_runtime.h>
#include <math.h>

typedef float v2f __attribute__((ext_vector_type(2)));
typedef float v8f __attribute__((ext_vector_type(8)));

#define LDS_STRIDE 18   // dwords per 16-float row: even (b64-aligned) and bank-conflict-free

// ---------------------------------------------------------------------------
// Kernel 1: build the batch-uniform 16x16 unitary for the weight part of the
// circuit (2 layers of RY/RZ per wire + CNOT ring), column per thread.
// Writes Re(U) to ws[0..255] and Im(U) to ws[256..511], row-major U[k][i]
// (psi_k = sum_i U[k][i] * s_i).
// ---------------------------------------------------------------------------
__global__ void build_unitary(const float* __restrict__ w, float* __restrict__ ws) {
    int j = threadIdx.x;
    if (j >= 16) return;
    float re[16], im[16];
    for (int i = 0; i < 16; ++i) { re[i] = (i == j) ? 1.f : 0.f; im[i] = 0.f; }

    for (int layer = 0; layer < 2; ++layer) {
        for (int q = 0; q < 4; ++q) {
            const int bit = 8 >> q;                      // wire q lives at bit (3-q)
            float hy = 0.5f * w[layer*8 + q*2 + 0];
            float cy = __cosf(hy), sy = __sinf(hy);
            for (int i = 0; i < 16; ++i) {
                if (i & bit) continue;
                int i1 = i | bit;
                float r0 = re[i], m0 = im[i], r1 = re[i1], m1 = im[i1];
                re[i]  = cy*r0 - sy*r1;  im[i]  = cy*m0 - sy*m1;
                re[i1] = sy*r0 + cy*r1;  im[i1] = sy*m0 + cy*m1;
            }
            float hz = 0.5f * w[layer*8 + q*2 + 1];
            float cz = __cosf(hz), sz = __sinf(hz);
            for (int i = 0; i < 16; ++i) {
                float r = re[i], m = im[i];
                if (i & bit) { re[i] = cz*r - sz*m; im[i] = cz*m + sz*r; }
                else         { re[i] = cz*r + sz*m; im[i] = cz*m - sz*r; }
            }
        }
        for (int q = 0; q < 4; ++q) {                    // CNOT(q, (q+1)%4)
            const int cbit = 8 >> q;
            const int tbit = 8 >> ((q + 1) & 3);
            for (int i = 0; i < 16; ++i) {
                if ((i & cbit) && !(i & tbit)) {
                    int i1 = i | tbit;
                    float tr = re[i]; re[i] = re[i1]; re[i1] = tr;
                    float tm = im[i]; im[i] = im[i1]; im[i1] = tm;
                }
            }
        }
    }
    for (int k = 0; k < 16; ++k) {
        ws[      k*16 + j] = re[k];
        ws[256 + k*16 + j] = im[k];
    }
}

// tanh via hardware EXP + RCP: tanh(x) = 1 - 2/(e^{2x}+1); exact at +-inf.
__device__ __forceinline__ float fast_tanh(float x) {
    float e = __expf(x + x);
    return 1.f - 2.f * __builtin_amdgcn_rcpf(e + 1.f);
}

// ---------------------------------------------------------------------------
// Kernel 2: per wave, 16-row batch tiles.
//   GEMM1: Yr/Yi = S @ U{r,i}^T   (8x v_wmma_f32_16x16x4_f32, A built in-reg)
//   p = Yr^2 + Yi^2 (D-layout), transposed to A-layout via same-wave LDS
//   GEMM2: out = P @ Zsign        (4x v_wmma, constant sign-matrix B)
//   D-layout of GEMM2 puts out[row][q] on lanes m<4 -> direct stores.
// ---------------------------------------------------------------------------
__global__ __launch_bounds__(256) void qcircuit(const float* __restrict__ x,
                                                const float* __restrict__ ws,
                                                float* __restrict__ out,
                                                int ntiles) {
    __shared__ float lds[8][16 * LDS_STRIDE];
    const int lane = threadIdx.x & 31;
    const int hi   = lane >> 4;       // K/M sub-half selector in WMMA layouts
    const int m    = lane & 15;       // A-row / B-column / D-column index
    float* lp = &lds[threadIdx.x >> 5][0];

    const int wave   = (blockIdx.x * blockDim.x + threadIdx.x) >> 5;
    const int nwaves = (gridDim.x * blockDim.x) >> 5;

    // GEMM1 B operands (batch-invariant): B[k_local][n] = U{r,i}[n][4kk+k_local]
    v2f br[4], bi[4];
#pragma unroll
    for (int kk = 0; kk < 4; ++kk) {
        const int col = 4*kk + 2*hi;
        br[kk] = *(const v2f*)(ws +       m*16 + col);
        bi[kk] = *(const v2f*)(ws + 256 + m*16 + col);
    }

    // GEMM2 B operand: Zs[k][n] = (n<4) ? (-1)^{bit_{3-n}(k)} : 0
    v2f bz[4];
    {
        const int sh = (m < 4) ? (3 - m) : 0;
#pragma unroll
        for (int kk = 0; kk < 4; ++kk) {
            const int k0 = 4*kk + 2*hi;
            bz[kk].x = (m < 4) ? (1.f - 2.f * (float)((k0       >> sh) & 1)) : 0.f;
            bz[kk].y = (m < 4) ? (1.f - 2.f * (float)(((k0 + 1) >> sh) & 1)) : 0.f;
        }
    }

    const float HALF_PI = 1.5707963267948966f;

    for (int t = wave; t < ntiles; t += nwaves) {
        // Angles for batch row (t*16 + m); lanes l and l+16 share a row.
        const float4 xv = *(const float4*)(x + (t*16 + m)*4);
        float c0,s0,c1,s1,c2,s2,c3,s3;
        { float h = fast_tanh(xv.x) * HALF_PI; c0 = __cosf(h); s0 = __sinf(h); }
        { float h = fast_tanh(xv.y) * HALF_PI; c1 = __cosf(h); s1 = __sinf(h); }
        { float h = fast_tanh(xv.z) * HALF_PI; c2 = __cosf(h); s2 = __sinf(h); }
        { float h = fast_tanh(xv.w) * HALF_PI; c3 = __cosf(h); s3 = __sinf(h); }
        const float w2 = hi ? s2 : c2;   // bit1 of state index = hi

        v8f yr = {0.f,0.f,0.f,0.f,0.f,0.f,0.f,0.f};
        v8f yi = {0.f,0.f,0.f,0.f,0.f,0.f,0.f,0.f};
#pragma unroll
        for (int kk = 0; kk < 4; ++kk) {
            // A fragment: S[m][k], k = 4kk + 2hi + {0,1};
            // bits: b3=kk>>1 (wire0), b2=kk&1 (wire1), b1=hi (wire2), b0 (wire3)
            const float w0 = (kk & 2) ? s0 : c0;
            const float w1 = (kk & 1) ? s1 : c1;
            const float base = w0 * w1 * w2;
            v2f a; a.x = base * c3; a.y = base * s3;
            yr = __builtin_amdgcn_wmma_f32_16x16x4_f32(false, a, false, br[kk],
                                                       (short)0, yr, false, false);
            yi = __builtin_amdgcn_wmma_f32_16x16x4_f32(false, a, false, bi[kk],
                                                       (short)0, yi, false, false);
        }

        // p (D-layout: row v+8hi, state m) -> LDS row-major [row][state]
#pragma unroll
        for (int v = 0; v < 8; ++v) {
            const float p = yr[v]*yr[v] + yi[v]*yi[v];
            lp[(v + 8*hi)*LDS_STRIDE + m] = p;
        }
        asm volatile("s_wait_dscnt 0x0" ::: "memory");   // same-wave LDS RAW fence

        v8f d2 = {0.f,0.f,0.f,0.f,0.f,0.f,0.f,0.f};
#pragma unroll
        for (int kk = 0; kk < 4; ++kk) {
            // A fragment of P: row m, k = 4kk + 2hi + {0,1}
            const v2f a2 = *(const v2f*)(lp + m*LDS_STRIDE + 4*kk + 2*hi);
            d2 = __builtin_amdgcn_wmma_f32_16x16x4_f32(false, a2, false, bz[kk],
                                                       (short)0, d2, false, false);
        }

        // D2[row=v+8hi][n=m] = out[t*16+row][q=m] for m<4
        if (m < 4) {
            float* o = out + (t*16 + 8*hi)*4 + m;
#pragma unroll
            for (int v = 0; v < 8; ++v) o[v*4] = d2[v];
        }
    }
}

extern "C" void kernel_launch(void* const* d_in, const int* in_sizes, int n_in,
                              void* d_out, int out_size, void* d_ws, size_t ws_size,
                              hipStream_t stream) {
    const float* x   = (const float*)d_in[0];   // [BATCH, 4] f32
    const float* w   = (const float*)d_in[1];   // [2, 4, 2]  f32
    float*       out = (float*)d_out;           // [BATCH, 4] f32
    float*       ws  = (float*)d_ws;            // 512 floats used

    build_unitary<<<1, 32, 0, stream>>>(w, ws);

    const int nrows  = in_sizes[0] / 4;         // BATCH
    const int ntiles = nrows / 16;              // 32768
    qcircuit<<<1024, 256, 0, stream>>>(x, ws, out, ntiles);
}